// GrouperBlockOld_57475252355560
// MI455X (gfx1250) — compile-verified
//
#include <hip/hip_runtime.h>
#include <math.h>

// Problem constants (from reference)
#define BATCH  8
#define NPTS   4096
#define NGROUP 1024
#define KNN    32
#define DPTS   64
#define DNMS   3
#define OUTC   (2*DPTS + 2*DNMS)   // 134

typedef __attribute__((ext_vector_type(2))) float v2f;
typedef __attribute__((ext_vector_type(8))) float v8f;

// ---------------------------------------------------------------------------
// Kernel 1: farthest point sampling. One workgroup per batch; xyz + running
// min-distance live in LDS (48KB + 16KB < 320KB WGP LDS). 1024 sequential
// argmax iterations with wave-shuffle + LDS tree reduction.
// ---------------------------------------------------------------------------
__global__ void fps_kernel(const float* __restrict__ xyz, int* __restrict__ fps)
{
    extern __shared__ float smem[];
    float* sxyz  = smem;                 // NPTS*3
    float* sdist = smem + NPTS*3;        // NPTS
    float* rv    = sdist + NPTS;         // 16 (per-wave max val)
    int*   ri    = (int*)(rv + 16);      // 16 (per-wave max idx)
    int*   sfar  = ri + 16;              // 1

    const int b   = blockIdx.x;
    const int tid = threadIdx.x;         // 512 threads = 16 waves
    const int lane = tid & 31, wid = tid >> 5;

    for (int i = tid; i < NPTS*3; i += 512) sxyz[i] = xyz[(size_t)b*NPTS*3 + i];
    for (int i = tid; i < NPTS;   i += 512) sdist[i] = 1e10f;
    __syncthreads();

    int far = 0;
    for (int t = 0; t < NGROUP; ++t) {
        if (tid == 0) fps[b*NGROUP + t] = far;
        const float cx = sxyz[far*3+0], cy = sxyz[far*3+1], cz = sxyz[far*3+2];

        float bv = -1.0f; int bi = 0;
        for (int n = tid; n < NPTS; n += 512) {
            float dx = sxyz[n*3+0]-cx, dy = sxyz[n*3+1]-cy, dz = sxyz[n*3+2]-cz;
            float d  = fminf(sdist[n], dx*dx+dy*dy+dz*dz);
            sdist[n] = d;
            if (d > bv) { bv = d; bi = n; }      // ascending n keeps earliest tie
        }
        // wave32 reduce: max value, tie -> lower index (matches jnp.argmax)
        #pragma unroll
        for (int off = 16; off; off >>= 1) {
            float ov = __shfl_xor(bv, off, 32);
            int   oi = __shfl_xor(bi, off, 32);
            if (ov > bv || (ov == bv && oi < bi)) { bv = ov; bi = oi; }
        }
        if (lane == 0) { rv[wid] = bv; ri[wid] = bi; }
        __syncthreads();
        if (tid < 32) {
            float v  = (tid < 16) ? rv[tid] : -2.0f;
            int   i2 = (tid < 16) ? ri[tid] : 0x7fffffff;
            #pragma unroll
            for (int off = 16; off; off >>= 1) {
                float ov = __shfl_xor(v,  off, 32);
                int   oi = __shfl_xor(i2, off, 32);
                if (ov > v || (ov == v && oi < i2)) { v = ov; i2 = oi; }
            }
            if (tid == 0) *sfar = i2;
        }
        __syncthreads();
        far = *sfar;
    }
}

// ---------------------------------------------------------------------------
// Kernel 2: build WMMA B-matrix columns: (-2x, -2y, -2z, |p|^2).
// ---------------------------------------------------------------------------
__global__ void prep_kernel(const float* __restrict__ xyz, float4* __restrict__ bmat)
{
    const int total = BATCH * NPTS;
    for (int i = blockIdx.x*blockDim.x + threadIdx.x; i < total;
         i += gridDim.x*blockDim.x) {
        float x = xyz[3*i+0], y = xyz[3*i+1], z = xyz[3*i+2];
        bmat[i] = make_float4(-2.0f*x, -2.0f*y, -2.0f*z, x*x + y*y + z*z);
    }
}

// ---------------------------------------------------------------------------
// Kernel 3: KNN via V_WMMA_F32_16X16X4_F32.
// Workgroup = (batch, 16-group tile). 8 waves each sweep disjoint 16-column
// tiles of the 16x4096 score slab (scores = -2 g.n + |n|^2; ranking-equal to
// the full sqr distance). Slab lives in LDS (256KB). Then per-row iterative
// min selection (tie -> lower index, matching lax.top_k).
// ---------------------------------------------------------------------------
__global__ void knn_kernel(const float* __restrict__ xyz,
                           const float4* __restrict__ bmat,
                           const int* __restrict__ fps,
                           int* __restrict__ knn)
{
    extern __shared__ float sdist[];     // 16 * NPTS floats = 256KB
    const int b    = blockIdx.x >> 6;
    const int g0   = (blockIdx.x & 63) << 4;
    const int tid  = threadIdx.x;        // 256 threads = 8 waves
    const int lane = tid & 31, wid = tid >> 5;
    const int m    = lane & 15;

    // A fragment: 16x4 f32, row m = sampled group point (x,y,z,1).
    // lanes 0-15 hold K=0,1 ; lanes 16-31 hold K=2,3 (ISA 32-bit A layout).
    const int gidx = fps[b*NGROUP + g0 + m];
    const float* p = xyz + ((size_t)b*NPTS + gidx)*3;
    v2f afrag;
    if (lane < 16) { afrag.x = p[0]; afrag.y = p[1]; }
    else           { afrag.x = p[2]; afrag.y = 1.0f; }

    const float4* bb = bmat + (size_t)b*NPTS;
    for (int t = wid; t < NPTS/16; t += 8) {
        const int n0 = t*16;
        float4 col = bb[n0 + m];
        if (n0 + 128 + m < NPTS)
            __builtin_prefetch(&bb[n0 + 128 + m], 0, 1);   // global_prefetch_b8
        // B fragment 4x16: lanes 0-15 hold K=0,1 ; lanes 16-31 hold K=2,3.
        v2f bfrag;
        if (lane < 16) { bfrag.x = col.x; bfrag.y = col.y; }
        else           { bfrag.x = col.z; bfrag.y = col.w; }
        v8f acc = {};
        acc = __builtin_amdgcn_wmma_f32_16x16x4_f32(
                  /*neg_a=*/false, afrag, /*neg_b=*/false, bfrag,
                  /*c_mod=*/(short)0, acc, /*reuse_a=*/false, /*reuse_b=*/false);
        // C/D layout: VGPR r -> M = r + 8*(lane>=16), N = lane%16
        const int ncol  = n0 + m;
        const int rbase = (lane < 16) ? 0 : 8;
        #pragma unroll
        for (int r = 0; r < 8; ++r)
            sdist[(rbase + r)*NPTS + ncol] = acc[r];
    }
    __syncthreads();

    // Top-32 smallest per row; wave w owns rows w and w+8.
    const float FINF = __int_as_float(0x7f800000);
    for (int rr = 0; rr < 2; ++rr) {
        const int row  = wid + rr*8;
        float* drow    = sdist + row*NPTS;
        const int gout = (b*NGROUP + g0 + row)*KNN;
        for (int j = 0; j < KNN; ++j) {
            float bv = FINF; int bi = NPTS;
            for (int n = lane; n < NPTS; n += 32) {
                float v = drow[n];
                if (v < bv) { bv = v; bi = n; }
            }
            #pragma unroll
            for (int off = 16; off; off >>= 1) {
                float ov = __shfl_xor(bv, off, 32);
                int   oi = __shfl_xor(bi, off, 32);
                if (ov < bv || (ov == bv && oi < bi)) { bv = ov; bi = oi; }
            }
            drow[bi] = FINF;                 // all lanes write same value/addr
            if (lane == 0) knn[gout + j] = bi;
        }
    }
}

// ---------------------------------------------------------------------------
// Kernel 4: zero per-batch accumulators (ws is not re-poisoned between calls).
// ---------------------------------------------------------------------------
__global__ void init_kernel(double* accum)
{
    if (threadIdx.x < 4*BATCH) accum[threadIdx.x] = 0.0;
}

// ---------------------------------------------------------------------------
// Kernel 5: per-batch sum / sumsq of (grouped - mean) diffs (pts & normals),
// f64 block-reduce then global f64 atomics.
// ---------------------------------------------------------------------------
__global__ void stats_kernel(const float* __restrict__ points,
                             const float* __restrict__ normals,
                             const int* __restrict__ fps,
                             const int* __restrict__ knn,
                             double* __restrict__ accum)
{
    const int b = blockIdx.x >> 10;
    const int g = blockIdx.x & 1023;
    const int tid = threadIdx.x;     // 256
    const int fidx = fps[b*NGROUP + g];
    const int* kid = knn + (size_t)(b*NGROUP + g)*KNN;

    double s1p = 0.0, s2p = 0.0, s1n = 0.0, s2n = 0.0;
    for (int e = tid; e < KNN*DPTS; e += 256) {
        const int k = e >> 6, d = e & 63;
        float diff = points[((size_t)b*NPTS + kid[k])*DPTS + d]
                   - points[((size_t)b*NPTS + fidx  )*DPTS + d];
        s1p += diff; s2p += (double)diff * diff;
    }
    for (int e = tid; e < KNN*DNMS; e += 256) {
        const int k = e / DNMS, d = e % DNMS;
        float diff = normals[((size_t)b*NPTS + kid[k])*DNMS + d]
                   - normals[((size_t)b*NPTS + fidx  )*DNMS + d];
        s1n += diff; s2n += (double)diff * diff;
    }

    __shared__ double rs[256];
    double vals[4] = { s1p, s2p, s1n, s2n };
    for (int q = 0; q < 4; ++q) {
        rs[tid] = vals[q]; __syncthreads();
        for (int s = 128; s > 0; s >>= 1) {
            if (tid < s) rs[tid] += rs[tid + s];
            __syncthreads();
        }
        if (tid == 0) atomicAdd(&accum[b*4 + q], rs[0]);   // global_atomic_add_f64
        __syncthreads();
    }
}

// ---------------------------------------------------------------------------
// Kernel 6: write output [gp(64) | mean(64) | gn(3) | nmean(3)] per (b,g,k).
// ---------------------------------------------------------------------------
__global__ void out_kernel(const float* __restrict__ points,
                           const float* __restrict__ normals,
                           const int* __restrict__ fps,
                           const int* __restrict__ knn,
                           const double* __restrict__ accum,
                           float* __restrict__ out)
{
    const int b = blockIdx.x >> 10;
    const int g = blockIdx.x & 1023;
    const int tid = threadIdx.x;     // 256
    const int fidx = fps[b*NGROUP + g];
    const int* kid = knn + (size_t)(b*NGROUP + g)*KNN;

    __shared__ float pmean[DPTS];
    __shared__ float nmean[DNMS];
    __shared__ float sstd[2];
    if (tid < DPTS) pmean[tid] = points[((size_t)b*NPTS + fidx)*DPTS + tid];
    if (tid >= DPTS && tid < DPTS+DNMS)
        nmean[tid-DPTS] = normals[((size_t)b*NPTS + fidx)*DNMS + (tid-DPTS)];
    if (tid == DPTS+DNMS) {
        const double Mp = (double)NGROUP*KNN*DPTS;
        const double Mn = (double)NGROUP*KNN*DNMS;
        double s1 = accum[b*4+0], s2 = accum[b*4+1];
        double t1 = accum[b*4+2], t2 = accum[b*4+3];
        sstd[0] = (float)sqrt(fmax(0.0, (s2 - s1*s1/Mp) / (Mp - 1.0)));
        sstd[1] = (float)sqrt(fmax(0.0, (t2 - t1*t1/Mn) / (Mn - 1.0)));
    }
    __syncthreads();

    const float pinv = 1.0f / (sstd[0] + 1e-5f);
    const float ninv = 1.0f / (sstd[1] + 1e-5f);
    float* orow = out + ((size_t)(b*NGROUP + g)*KNN)*OUTC;
    for (int e = tid; e < KNN*OUTC; e += 256) {
        const int k = e / OUTC, c = e % OUTC;
        const int idx = kid[k];
        float val;
        if (c < DPTS)
            val = (points[((size_t)b*NPTS + idx)*DPTS + c] - pmean[c]) * pinv;
        else if (c < 2*DPTS)
            val = pmean[c - DPTS];
        else if (c < 2*DPTS + DNMS)
            val = (normals[((size_t)b*NPTS + idx)*DNMS + (c - 2*DPTS)] - nmean[c - 2*DPTS]) * ninv;
        else
            val = nmean[c - 2*DPTS - DNMS];
        orow[(size_t)k*OUTC + c] = val;
    }
}

// ---------------------------------------------------------------------------
// Host launcher.  Workspace layout (bytes):
//   [0,        32768)   fps_idx   : B*G int32
//   [32768,   557056)   bmat      : B*N float4
//   [557056, 1605632)   knn_idx   : B*G*K int32
//   [1605632,1605888)   accum     : B*4 double
// ---------------------------------------------------------------------------
extern "C" void kernel_launch(void* const* d_in, const int* in_sizes, int n_in,
                              void* d_out, int out_size, void* d_ws, size_t ws_size,
                              hipStream_t stream)
{
    const float* xyz     = (const float*)d_in[0];
    // d_in[1] (normal_xyz) is unused by the reference
    const float* normals = (const float*)d_in[2];
    const float* points  = (const float*)d_in[3];
    float* out = (float*)d_out;

    char* ws = (char*)d_ws;
    int*    fps   = (int*)   (ws + 0);
    float4* bmat  = (float4*)(ws + 32768);
    int*    knn   = (int*)   (ws + 557056);
    double* accum = (double*)(ws + 1605632);

    const size_t fps_smem = (size_t)NPTS*3*4 + (size_t)NPTS*4 + 16*4 + 16*4 + 4;
    const size_t knn_smem = (size_t)16*NPTS*4;        // 256KB (< 320KB WGP LDS)
    (void)hipFuncSetAttribute((const void*)fps_kernel,
        hipFuncAttributeMaxDynamicSharedMemorySize, (int)fps_smem);
    (void)hipFuncSetAttribute((const void*)knn_kernel,
        hipFuncAttributeMaxDynamicSharedMemorySize, (int)knn_smem);

    fps_kernel <<<BATCH,            512, fps_smem, stream>>>(xyz, fps);
    prep_kernel<<<128,              256, 0,        stream>>>(xyz, bmat);
    knn_kernel <<<BATCH*(NGROUP/16),256, knn_smem, stream>>>(xyz, bmat, fps, knn);
    init_kernel<<<1,                 64, 0,        stream>>>(accum);
    stats_kernel<<<BATCH*NGROUP,    256, 0,        stream>>>(points, normals, fps, knn, accum);
    out_kernel <<<BATCH*NGROUP,     256, 0,        stream>>>(points, normals, fps, knn, accum, out);
}